// CrossEntropy_84516366450885
// MI455X (gfx1250) — compile-verified
//
#include <hip/hip_runtime.h>

// Problem shape (fixed by the reference)
#define T_DIM 64
#define B_DIM 128
#define V_DIM 8192
#define ROWS (T_DIM * B_DIM)   // 8192 rows of length V
#define NTHR 256               // 8 waves (wave32)
#define TOPK 20u
#define LOG_EPS 1e-8f

typedef __attribute__((ext_vector_type(2))) float v2f;
typedef __attribute__((ext_vector_type(8))) float v8f;
typedef __attribute__((ext_vector_type(4))) unsigned v4u;
typedef __attribute__((ext_vector_type(8))) unsigned v8u;

// One workgroup per (t,b) row.
//  - TDM (tensor_load_to_lds) DMAs the 32KB y_hat row into LDS while the
//    waves stream the y row from HBM and build a positive bitmask in LDS
//  - CE pass: one __logf per element (y is binary -> select log argument)
//  - per-wave CE sums reduced with V_WMMA_F32_16X16X4_F32 (ones-A trick)
//  - 4-round radix select on float bit patterns -> 20th-largest threshold
//  - TP = positives with value >= threshold
__global__ __launch_bounds__(NTHR) void bce_topk_rows(
    const float* __restrict__ y_hat, const float* __restrict__ y,
    const float* __restrict__ length, float* __restrict__ ws_ce,
    float* __restrict__ ws_tp, float* __restrict__ ws_np) {
  __shared__ __align__(16) float s_p[V_DIM];  // 32 KB DMA-staged y_hat row
  __shared__ unsigned s_mask[V_DIM / 32];     // 1 KB positive bitmask
  __shared__ unsigned s_hist[256];            // radix histogram
  __shared__ float    s_wsum[8];              // per-wave CE sums
  __shared__ int      s_red[2];               // tp, npos
  __shared__ unsigned s_sel[2];               // prefix, k-remaining

  const int row  = blockIdx.x;          // row = t*B + b
  const int b    = row & (B_DIM - 1);
  const int tid  = threadIdx.x;
  const int lane = tid & 31;
  const int wave = tid >> 5;

  s_mask[tid] = 0u;
  if (tid < 2) s_red[tid] = 0;
  if (tid == 0) { s_sel[0] = 0u; s_sel[1] = TOPK; }

  // ---- wave 0: kick off the TDM DMA of this row's y_hat into LDS ----
  if (tid < 32) {
    const unsigned long long ga =
        (unsigned long long)(uintptr_t)(y_hat + (size_t)row * V_DIM);
    const unsigned lds_addr = (unsigned)(uintptr_t)(&s_p[0]);  // low 32 bits = LDS offset
    v4u g0;
    g0[0] = 1u;                                   // count=1, user mode, no gather
    g0[1] = lds_addr;                             // lds_addr (bytes)
    g0[2] = (unsigned)(ga & 0xFFFFFFFFu);         // global_addr[31:0]
    g0[3] = (unsigned)((ga >> 32) & 0x01FFFFFFu)  // global_addr[56:32]
            | (2u << 30);                         // type = 2 ("image")
    v8u g1;
    g1[0] = 2u << 16;                             // data_size = 4 bytes; no mcast/pad
    g1[1] = (V_DIM & 0xFFFFu) << 16;              // tensor_dim0[15:0] = 8192
    g1[2] = 1u << 16;                             // tensor_dim0[31:16]=0, tensor_dim1=1
    g1[3] = (V_DIM & 0xFFFFu) << 16;              // tile_dim0 = 8192
    g1[4] = 1u;                                   // tile_dim1 = 1, tile_dim2 = 0
    g1[5] = (unsigned)V_DIM;                      // tensor_dim0_stride[31:0] = 8192
    g1[6] = (V_DIM & 0xFFFFu) << 16;              // dim0_stride[47:32]=0, dim1_stride lo
    g1[7] = 0u;                                   // dim1_stride hi
    asm volatile("tensor_load_to_lds %0, %1" : : "s"(g0), "s"(g1) : "memory");
  }

  __syncthreads();  // publish s_mask/s_red init before atomics below

  // ---- stream y row (coalesced b128) while the TDM runs: mask + npos ----
  const float4* yrow4 =
      reinterpret_cast<const float4*>(y) + (size_t)row * (V_DIM / 4);
  int npos_loc = 0;
#pragma unroll
  for (int i = 0; i < V_DIM / 4 / NTHR; ++i) {  // 8 iterations
    const int idx4 = i * NTHR + tid;
    __builtin_prefetch(yrow4 + idx4 + NTHR, 0, 0);  // global_prefetch_b8
    const float4 t = yrow4[idx4];
    const unsigned nib = (t.x != 0.f ? 1u : 0u) | (t.y != 0.f ? 2u : 0u) |
                         (t.z != 0.f ? 4u : 0u) | (t.w != 0.f ? 8u : 0u);
    npos_loc += __popc(nib);
    if (nib) {
      const int e0 = idx4 * 4;
      atomicOr(&s_mask[e0 >> 5], nib << (e0 & 31));
    }
  }

  // ---- wait for the DMA, then make it visible to all waves ----
  if (tid < 32) __builtin_amdgcn_s_wait_tensorcnt(0);
  __syncthreads();

  // ---- CE pass from LDS: y binary -> ONE log per element ----
  const float4* s_p4 = reinterpret_cast<const float4*>(s_p);
  float ce0 = 0.f, ce1 = 0.f;
#pragma unroll
  for (int i = 0; i < V_DIM / 4 / NTHR; ++i) {
    const int idx4 = i * NTHR + tid;
    const float4 p = s_p4[idx4];
    const int e0 = idx4 * 4;
    const unsigned nib = (s_mask[e0 >> 5] >> (e0 & 31)) & 0xFu;
    ce0 -= __logf((nib & 1u) ? (p.x + LOG_EPS) : (1.f - p.x + LOG_EPS));
    ce1 -= __logf((nib & 2u) ? (p.y + LOG_EPS) : (1.f - p.y + LOG_EPS));
    ce0 -= __logf((nib & 4u) ? (p.z + LOG_EPS) : (1.f - p.z + LOG_EPS));
    ce1 -= __logf((nib & 8u) ? (p.w + LOG_EPS) : (1.f - p.w + LOG_EPS));
  }

  // Per-wave CE reduction via WMMA: D = ones(16x4) * B(4x16) + 0.
  // With A == all-ones, every D entry is a column-sum of B, so
  // sum(D row) == sum of all 64 operand values, independent of lane layout.
  v2f a_ones; a_ones.x = 1.0f; a_ones.y = 1.0f;
  v2f b_dat;  b_dat.x  = ce0;  b_dat.y  = ce1;
  v8f acc = {};
  acc = __builtin_amdgcn_wmma_f32_16x16x4_f32(false, a_ones, false, b_dat,
                                              (short)0, acc, false, false);
  float wsum = acc[0];  // lanes 0-15: D[0][n], lanes 16-31: D[8][n] (identical rows)
#pragma unroll
  for (int off = 16; off >= 1; off >>= 1) wsum += __shfl_xor(wsum, off, 32);
  wsum *= 0.5f;  // rows 0 and 8 each contributed the full total
  if (lane == 0) s_wsum[wave] = wsum;
  __syncthreads();

  // Radix select (MSB-first, 4 bytes) for the bit pattern of the 20th largest.
  // Values are in (0,1): positive floats, so uint compare == float compare.
#pragma unroll 1
  for (int round = 0; round < 4; ++round) {
    const int shift = 24 - 8 * round;
    const unsigned pmask  = (round == 0) ? 0u : (0xFFFFFFFFu << (shift + 8));
    const unsigned prefix = s_sel[0];
    s_hist[tid] = 0u;
    __syncthreads();
    for (int i = 0; i < V_DIM / NTHR; ++i) {
      const int e = i * NTHR + tid;
      const unsigned u = __float_as_uint(s_p[e]);
      if ((u & pmask) == prefix) atomicAdd(&s_hist[(u >> shift) & 0xFFu], 1u);
    }
    __syncthreads();
    if (tid == 0) {
      unsigned k = s_sel[1], cum = 0u, bsel = 0u;
      for (int bkt = 255; bkt >= 0; --bkt) {
        const unsigned c = s_hist[bkt];
        if (cum + c >= k) { bsel = (unsigned)bkt; s_sel[1] = k - cum; break; }
        cum += c;
      }
      s_sel[0] = prefix | (bsel << shift);
    }
    __syncthreads();
  }
  const unsigned thr = s_sel[0];  // bit pattern of the 20th-largest value

  // Count positives inside the top-20 set.
  int tp_loc = 0;
  for (int i = 0; i < V_DIM / NTHR; ++i) {
    const int e = i * NTHR + tid;
    const unsigned u = __float_as_uint(s_p[e]);
    if (u >= thr && ((s_mask[e >> 5] >> (e & 31)) & 1u)) tp_loc++;
  }
  if (tp_loc)   atomicAdd(&s_red[0], tp_loc);
  if (npos_loc) atomicAdd(&s_red[1], npos_loc);
  __syncthreads();

  if (tid == 0) {
    float bsum = 0.f;
#pragma unroll
    for (int w = 0; w < 8; ++w) bsum += s_wsum[w];
    ws_ce[row] = bsum / (length[b] * (float)B_DIM);  // cost = sum of these
    ws_tp[row] = (float)s_red[0];
    ws_np[row] = (float)s_red[1];
  }
}

// Single-block deterministic reduction of the 8192 per-row partials.
__global__ __launch_bounds__(NTHR) void finalize_k(
    const float* __restrict__ ws_ce, const float* __restrict__ ws_tp,
    const float* __restrict__ ws_np, float* __restrict__ out) {
  __shared__ float sc[NTHR], st[NTHR], sn[NTHR];
  const int tid = threadIdx.x;
  float c = 0.f, t = 0.f, n = 0.f;
  for (int i = tid; i < ROWS; i += NTHR) {
    c += ws_ce[i];
    t += ws_tp[i];
    n += ws_np[i];
  }
  sc[tid] = c; st[tid] = t; sn[tid] = n;
  __syncthreads();
  for (int s = NTHR / 2; s > 0; s >>= 1) {
    if (tid < s) { sc[tid] += sc[tid + s]; st[tid] += st[tid + s]; sn[tid] += sn[tid + s]; }
    __syncthreads();
  }
  if (tid == 0) {
    out[0] = sc[0];                    // cost
    out[1] = st[0] / (sn[0] + 1.0f);   // top-20 accuracy
  }
}

extern "C" void kernel_launch(void* const* d_in, const int* in_sizes, int n_in,
                              void* d_out, int out_size, void* d_ws, size_t ws_size,
                              hipStream_t stream) {
  (void)in_sizes; (void)n_in; (void)out_size; (void)ws_size;
  const float* y_hat  = (const float*)d_in[0];  // [T,B,V] fp32
  const float* y      = (const float*)d_in[1];  // [T,B,V] fp32
  const float* length = (const float*)d_in[2];  // [B] fp32
  float* out = (float*)d_out;                   // [cost, acc]

  float* ws    = (float*)d_ws;                  // 3*ROWS floats = 96 KB scratch
  float* ws_ce = ws;
  float* ws_tp = ws + ROWS;
  float* ws_np = ws + 2 * ROWS;

  bce_topk_rows<<<ROWS, NTHR, 0, stream>>>(y_hat, y, length, ws_ce, ws_tp, ws_np);
  finalize_k<<<1, NTHR, 0, stream>>>(ws_ce, ws_tp, ws_np, out);
}